// UnSupervisedLoss_4045859193149
// MI455X (gfx1250) — compile-verified
//
#include <hip/hip_runtime.h>
#include <hip/hip_bf16.h>

typedef __attribute__((ext_vector_type(2))) float v2f;
typedef __attribute__((ext_vector_type(8))) float v8f;

#define DIST_THRESHOLD 3.33f
#define B_ 4
#define N_ 8192
#define TILES_ (N_ / 16)

// One wave per block. Wave owns a 16-point query tile for (dir, batch) and
// scans all 16-point search tiles, computing the 16x16 squared-distance tile
// with a single V_WMMA_F32_16X16X4_F32 per search tile:
//   A row  = (qx, qy, qz, |q|^2)     (K=4, 3-D point padded)
//   B col  = (-2px, -2py, -2pz, 1)
//   C      = |p|^2 broadcast per column
//   D      = |q|^2 - 2 q.p + |p|^2   (exact squared distance tile)
__global__ __launch_bounds__(32)
void chamfer_nn_min_kernel(const float* __restrict__ pts_src,
                           const float* __restrict__ pts_dst,
                           const float* __restrict__ flows,
                           const int* __restrict__ msk_src,
                           const int* __restrict__ msk_dst,
                           float* __restrict__ out_min)  // [2][B][N]
{
  const int lane = threadIdx.x;   // 0..31
  const int half = lane >> 4;     // lane group: 0 -> K0/K1 & rows 0..7, 1 -> K2/K3 & rows 8..15
  const int lid  = lane & 15;

  int blk = blockIdx.x;           // 0 .. 2*B*TILES-1
  const int dir = (blk >= B_ * TILES_) ? 1 : 0;
  if (dir) blk -= B_ * TILES_;
  const int b  = blk / TILES_;
  const int tq = blk % TILES_;

  // ---- load this wave's 16 query points, build WMMA A operand ----
  // 32-bit A 16x4 layout: lanes 0-15 hold (K0,K1), lanes 16-31 hold (K2,K3), M = lane&15
  const int    qi   = tq * 16 + lid;
  const size_t qoff = ((size_t)b * N_ + qi) * 3;
  float qx, qy, qz;
  if (dir == 0) {  // queries = warped src
    qx = pts_src[qoff + 0] + flows[qoff + 0];
    qy = pts_src[qoff + 1] + flows[qoff + 1];
    qz = pts_src[qoff + 2] + flows[qoff + 2];
  } else {         // queries = dst
    qx = pts_dst[qoff + 0];
    qy = pts_dst[qoff + 1];
    qz = pts_dst[qoff + 2];
  }
  const float q2 = qx * qx + qy * qy + qz * qz;
  v2f amat;
  amat.x = half ? qz : qx;
  amat.y = half ? q2 : qy;

  const float* __restrict__ spts = (dir == 0) ? pts_dst : pts_src;
  const int*   __restrict__ smsk = (dir == 0) ? msk_dst : msk_src;

  const float INF = __builtin_inff();
  float runmin[8];
#pragma unroll
  for (int r = 0; r < 8; ++r) runmin[r] = INF;

  for (int tp = 0; tp < TILES_; ++tp) {
    const int    pj   = tp * 16 + lid;
    const size_t poff = ((size_t)b * N_ + pj) * 3;
    float px = spts[poff + 0];
    float py = spts[poff + 1];
    float pz = spts[poff + 2];
    if (dir == 1) {  // search set = warped src
      px += flows[poff + 0];
      py += flows[poff + 1];
      pz += flows[poff + 2];
    }
    const float p2     = px * px + py * py + pz * pz;
    const bool  pvalid = smsk[(size_t)b * N_ + pj] > 0;

    // 32-bit B 4x16 layout: lanes 0-15 hold (K0,K1), lanes 16-31 hold (K2,K3), N = lane&15
    v2f bmat;
    bmat.x = half ? (-2.0f * pz) : (-2.0f * px);
    bmat.y = half ? 1.0f         : (-2.0f * py);

    v8f cmat;
#pragma unroll
    for (int r = 0; r < 8; ++r) cmat[r] = p2;  // fold +|p|^2 in via C

    v8f d = __builtin_amdgcn_wmma_f32_16x16x4_f32(
        false, amat, false, bmat, (short)0, cmat, false, false);

#pragma unroll
    for (int r = 0; r < 8; ++r) {
      const float dv = pvalid ? fmaxf(d[r], 0.0f) : INF;  // clamp, mask invalid search pts
      runmin[r] = fminf(runmin[r], dv);
    }
  }

  // Row-min across the 16 lanes (columns) of each half-wave.
  // C/D layout: VGPR r holds row r (lanes 0-15) or row 8+r (lanes 16-31).
#pragma unroll
  for (int r = 0; r < 8; ++r) {
    float v = runmin[r];
    v = fminf(v, __shfl_xor(v, 1, 16));
    v = fminf(v, __shfl_xor(v, 2, 16));
    v = fminf(v, __shfl_xor(v, 4, 16));
    v = fminf(v, __shfl_xor(v, 8, 16));
    runmin[r] = v;
  }

  if (lid == 0) {  // lanes 0 and 16 each write their 8 row-mins
    float* outp = out_min + ((size_t)dir * B_ + b) * N_ + tq * 16 + half * 8;
#pragma unroll
    for (int r = 0; r < 8; ++r) outp[r] = runmin[r];
  }
}

// Single block: masked, thresholded per-batch means in both directions, summed.
__global__ __launch_bounds__(256)
void chamfer_finalize_kernel(const float* __restrict__ mins,  // [2][B][N]
                             const int* __restrict__ msk_src,
                             const int* __restrict__ msk_dst,
                             float* __restrict__ out)
{
  __shared__ float s_sum[2][B_];
  __shared__ int   s_cnt[2][B_];
  if (threadIdx.x < 2 * B_) {
    (&s_sum[0][0])[threadIdx.x] = 0.0f;
    (&s_cnt[0][0])[threadIdx.x] = 0;
  }
  __syncthreads();

  float lsum[2][B_] = {};
  int   lcnt[2][B_] = {};
  for (int i = threadIdx.x; i < B_ * N_; i += blockDim.x) {
    const int   b  = i / N_;
    const float v0 = mins[i];                  // dir 0: per-src NN dist
    if (msk_src[i] > 0 && v0 < DIST_THRESHOLD) { lsum[0][b] += v0; lcnt[0][b] += 1; }
    const float v1 = mins[B_ * N_ + i];        // dir 1: per-dst NN dist
    if (msk_dst[i] > 0 && v1 < DIST_THRESHOLD) { lsum[1][b] += v1; lcnt[1][b] += 1; }
  }
#pragma unroll
  for (int d = 0; d < 2; ++d)
#pragma unroll
    for (int b = 0; b < B_; ++b) {
      atomicAdd(&s_sum[d][b], lsum[d][b]);
      atomicAdd(&s_cnt[d][b], lcnt[d][b]);
    }
  __syncthreads();

  if (threadIdx.x == 0) {
    float tot = 0.0f;
#pragma unroll
    for (int b = 0; b < B_; ++b)
      tot += s_sum[0][b] / (float)s_cnt[0][b] + s_sum[1][b] / (float)s_cnt[1][b];
    out[0] = tot;
  }
}

extern "C" void kernel_launch(void* const* d_in, const int* in_sizes, int n_in,
                              void* d_out, int out_size, void* d_ws, size_t ws_size,
                              hipStream_t stream) {
  const float* pts_src = (const float*)d_in[0];  // [B,N,3]
  const float* pts_dst = (const float*)d_in[1];  // [B,N,3]
  const float* flows   = (const float*)d_in[2];  // [B,N,3]
  // d_in[3] = flows_gt: unused by the reference loss
  const int* msk_src = (const int*)d_in[4];      // [B,N]
  const int* msk_dst = (const int*)d_in[5];      // [B,N]

  float* mins = (float*)d_ws;  // 2 * B * N floats = 256 KB scratch; fully overwritten

  chamfer_nn_min_kernel<<<2 * B_ * TILES_, 32, 0, stream>>>(
      pts_src, pts_dst, flows, msk_src, msk_dst, mins);
  chamfer_finalize_kernel<<<1, 256, 0, stream>>>(
      mins, msk_src, msk_dst, (float*)d_out);
}